// GATIDConv_28793460752469
// MI455X (gfx1250) — compile-verified
//
#include <hip/hip_runtime.h>
#include <hip/hip_bf16.h>
#include <stdint.h>

// ---------------------------------------------------------------------------
// GAT + per-label identity transform for MI455X (gfx1250, wave32).
//
// Phase 1 (dense, WMMA + async data movement):
//   ox = x + x @ W_k per label bucket, fp32 V_WMMA_F32_16X16X4_F32.
//   - x rows gathered into LDS with GLOBAL_LOAD_ASYNC_TO_LDS_B128 (per-lane
//     scattered addresses, ASYNCcnt).
//   - W staged per 128x32 chunk by the Tensor Data Mover (tensor_load_to_lds,
//     TENSORcnt), with D# LDS-padding producing a conflict-free stride-34
//     layout. K-loop is then pure LDS reads + WMMA (no global waits).
// Phase 2 (edge, memory-bound): alpha_e = a_i[dst] + a_j[src] via per-node
//   attention dots, segment-max with ordered-uint atomicMax, exp/segment-sum,
//   out[dst] += coeff * ox[src] with global_atomic_add_f32. ox+out = 51 MB
//   << 192 MB L2, so all gathers/atomics stay on-chip.
// ---------------------------------------------------------------------------

#define D           128
#define NUM_LABELS  7
#define NEG_SLOPE   0.2f

typedef __attribute__((ext_vector_type(2))) float    v2f;
typedef __attribute__((ext_vector_type(8))) float    v8f;
typedef __attribute__((ext_vector_type(4))) unsigned v4u;
typedef __attribute__((ext_vector_type(8))) unsigned v8u;

// monotone float <-> uint mapping so unsigned atomicMax == float max
__device__ __forceinline__ unsigned f2ord(float f) {
    unsigned u = __float_as_uint(f);
    return (u & 0x80000000u) ? ~u : (u | 0x80000000u);
}
__device__ __forceinline__ float ord2f(unsigned k) {
    unsigned u = (k & 0x80000000u) ? (k ^ 0x80000000u) : ~k;
    return __uint_as_float(u);
}

// low 32 bits of a generic pointer to LDS == LDS byte offset
__device__ __forceinline__ unsigned lds_off(const void* p) {
    return (unsigned)(uintptr_t)p;
}

// ---------------------------------------------------------------------------
// K0: init: ox = x, out = 0, amax keys = 0 (== -inf encoded), denom = 0,
// bucket counters = 0. Deterministic re-init every launch (graph replay safe).
// ---------------------------------------------------------------------------
__global__ void k_init(const float* __restrict__ x, float* __restrict__ ox,
                       float* __restrict__ out, unsigned* __restrict__ amaxkey,
                       float* __restrict__ denom, int* __restrict__ cnt, int N) {
    long long i = (long long)blockIdx.x * blockDim.x + threadIdx.x;
    long long total = (long long)N * D;
    if (i < total) { ox[i] = x[i]; out[i] = 0.0f; }
    if (i < N) { amaxkey[i] = 0u; denom[i] = 0.0f; }
    if (i < NUM_LABELS + 1) cnt[i] = 0;
}

// ---------------------------------------------------------------------------
// K0b: transpose weights once: Wt[k][o][d] = W[k][d][o]  (458 KB, L2-hot).
// B fragments need two K-adjacent values at a fixed output column -> make the
// K dimension contiguous so the TDM tile rows are contiguous runs.
// ---------------------------------------------------------------------------
__global__ void k_wtrans(const float* __restrict__ W, float* __restrict__ Wt) {
    int i = blockIdx.x * blockDim.x + threadIdx.x;   // over 7*128*128
    if (i >= NUM_LABELS * D * D) return;
    int k = i / (D * D);
    int o = (i / D) % D;
    int d = i % D;
    Wt[i] = W[(long long)k * D * D + (long long)d * D + o];
}

// ---------------------------------------------------------------------------
// K1: bucket node indices by label (labels 1..7 active; 0 inactive -> ox = x).
// ---------------------------------------------------------------------------
__global__ void k_bucket(const int* __restrict__ lab, int* __restrict__ cnt,
                         int* __restrict__ lists, int N) {
    int n = blockIdx.x * blockDim.x + threadIdx.x;
    if (n >= N) return;
    int l = lab[n];
    if (l >= 1 && l <= NUM_LABELS) {
        int k = l - 1;
        int pos = atomicAdd(&cnt[k], 1);
        lists[(long long)k * N + pos] = n;
    }
}

// ---------------------------------------------------------------------------
// K2: per-label GEMM with row gather, fp32 WMMA, async/TDM staging.
// Block = 128 threads (4 wave32s). Tile = 32 gathered rows x 128 cols.
// Wave w: row-tile (w&1), col-tiles 4*(w>>1)..+3  -> 4 x v8f accumulators.
// A (16x4 fp32) frag: lanes 0-15 hold K=k0,k0+1; lanes 16-31 K=k0+2,k0+3.
// B (4x16 fp32) frag: same K split, from the LDS-staged Wt chunk.
// xs row stride 132 floats, bs row stride 34 floats: both give 16 distinct
// banks for the 16-lane same-K reads and keep v2f reads 8B-aligned.
// ---------------------------------------------------------------------------
__global__ void k_gemm(const float* __restrict__ x, const float* __restrict__ Wt,
                       const int* __restrict__ cnt, const int* __restrict__ lists,
                       float* __restrict__ ox, int N) {
    const int k = blockIdx.y;                // label bucket
    const int count = cnt[k];
    const int tile0 = blockIdx.x * 32;
    if (tile0 >= count) return;              // uniform early-exit
    const int rcount = min(32, count - tile0);

    __shared__ float xs[32 * 132];           // 16.9 KB x-tile
    __shared__ float bs[128 * 34];           // 17.4 KB Wt chunk (TDM-padded)
    __shared__ int   nid[32];

    const int t = threadIdx.x;
    if (t < 32) nid[t] = lists[(long long)k * N + tile0 + min(t, rcount - 1)];
    __syncthreads();

    // -------- async gather: 32 scattered x rows -> LDS (ASYNCcnt) ----------
    // 1024 16-byte transfers across 128 lanes; per-lane global + LDS address.
#pragma unroll
    for (int i = 0; i < 8; ++i) {
        int f4  = i * 128 + t;               // 0..1023
        int row = f4 >> 5;
        int c4  = f4 & 31;
        unsigned lo = lds_off(xs + row * 132 + c4 * 4);            // 16B aligned
        uint64_t ga = (uint64_t)(uintptr_t)(x + (long long)nid[row] * D + c4 * 4);
        asm volatile("global_load_async_to_lds_b128 %0, %1, off"
                     :: "v"(lo), "v"(ga) : "memory");
    }
    asm volatile("s_wait_asynccnt 0x0" ::: "memory");
    __syncthreads();

    const int wave    = t >> 5;
    const int lane    = t & 31;
    const int rt      = wave & 1;            // row tile 0/1
    const int cbase   = (wave >> 1) * 4;     // col tiles cbase..cbase+3
    const int m       = lane & 15;
    const int kk      = (lane >> 4) * 2;     // K sub-offset per half-wave

    const float* __restrict__ Wtk = Wt + (long long)k * D * D;  // Wt[k][o][d]

    // ---- TDM descriptor (uniform), group1 is chunk-invariant ----
    // 2-D tile: 128 rows (output cols) x 32 K-values, 4B elements,
    // row stride in memory = 128 elements; LDS pad 2 DWORDs every 32 DWORDs
    // -> stride-34 rows in bs.
    v8u g1;
    g1[0] = (2u << 16)        // data_size = 4 bytes
          | (1u << 20)        // pad_enable
          | (4u << 22)        // pad_interval: 32 DWORDs
          | (1u << 25);       // pad_amount: 2 DWORDs
    g1[1] = (32u & 0xFFFFu) << 16;                    // tensor_dim0 = 32 (lo16)
    g1[2] = ((128u & 0xFFFFu) << 16);                 // dim0 hi=0 | tensor_dim1 = 128
    g1[3] = (32u << 16);                              // dim1 hi=0 | tile_dim0 = 32
    g1[4] = 128u;                                     // tile_dim1 = 128, tile_dim2 = 0
    g1[5] = 128u;                                     // tensor_dim0_stride = 128 (lo32)
    g1[6] = 0u;
    g1[7] = 0u;
    const unsigned bs_lo = lds_off(bs);

    v8f acc[4] = {v8f{}, v8f{}, v8f{}, v8f{}};

    for (int c = 0; c < 4; ++c) {            // four 32-K chunks
        // ---- stage Wt[k][:, c*32 .. c*32+32) into bs via TDM ----
        if (wave == 0) {
            uint64_t ga = (uint64_t)(uintptr_t)Wtk + (uint64_t)c * 32u * 4u;
            v4u g0;
            g0[0] = 1u;                                      // count = 1 (valid)
            g0[1] = bs_lo;                                   // lds_addr
            g0[2] = (unsigned)(ga & 0xFFFFFFFFu);            // global_addr lo
            g0[3] = (unsigned)((ga >> 32) & 0x01FFFFFFu)     // global_addr hi
                  | (2u << 30);                              // type = 2 (image)
            asm volatile("tensor_load_to_lds %0, %1\n\t"
                         "s_wait_tensorcnt 0x0"
                         :: "s"(g0), "s"(g1) : "memory");
        }
        __syncthreads();

        // ---- pure LDS + WMMA K-loop over this chunk ----
        for (int k0 = 0; k0 < 32; k0 += 4) {
            const v2f a = *reinterpret_cast<const v2f*>(
                xs + (rt * 16 + m) * 132 + c * 32 + k0 + kk);
#pragma unroll
            for (int j = 0; j < 4; ++j) {
                const int n0 = (cbase + j) * 16 + m;         // output column
                const v2f b = *reinterpret_cast<const v2f*>(
                    bs + n0 * 34 + k0 + kk);
                acc[j] = __builtin_amdgcn_wmma_f32_16x16x4_f32(
                    false, a, false, b, (short)0, acc[j], false, false);
            }
        }
        __syncthreads();   // bs reused next chunk
    }

    // ox = x + delta accumulated in LDS (disjoint per-wave regions,
    // bank-conflict-free: 16-consecutive-column runs).
    const int halfsel = lane >> 4;
#pragma unroll
    for (int j = 0; j < 4; ++j) {
        const int col = (cbase + j) * 16 + m;
#pragma unroll
        for (int r = 0; r < 8; ++r) {
            int rowInTile = rt * 16 + r + 8 * halfsel;
            xs[rowInTile * 132 + col] += acc[j][r];
        }
    }
    __syncthreads();

    // coalesced float4 store sweep (mirror of the gather sweep)
    for (int i = 0; i < 8; ++i) {
        int f4  = i * 128 + t;
        int row = f4 >> 5;
        int c4  = f4 & 31;
        if (row < rcount) {
            const float* sp = xs + row * 132 + c4 * 4;
            float4 v;
            v.x = sp[0]; v.y = sp[1]; v.z = sp[2]; v.w = sp[3];
            *reinterpret_cast<float4*>(ox + (long long)nid[row] * D + c4 * 4) = v;
        }
    }
}

// ---------------------------------------------------------------------------
// K3: per-node attention dots: a_i[n] = ox[n].att[:128], a_j[n] = ox[n].att[128:]
// One wave32 per node; one b128 load per lane; shuffle tree reduction.
// ---------------------------------------------------------------------------
__global__ void k_dots(const float* __restrict__ ox, const float* __restrict__ att,
                       float* __restrict__ a_i, float* __restrict__ a_j, int N) {
    int g    = (int)(((long long)blockIdx.x * blockDim.x + threadIdx.x) >> 5);
    int lane = threadIdx.x & 31;
    if (g >= N) return;
    const float4 v   = reinterpret_cast<const float4*>(ox + (long long)g * D)[lane];
    const float4 ai4 = reinterpret_cast<const float4*>(att)[lane];
    const float4 aj4 = reinterpret_cast<const float4*>(att)[32 + lane];
    float si = v.x * ai4.x + v.y * ai4.y + v.z * ai4.z + v.w * ai4.w;
    float sj = v.x * aj4.x + v.y * aj4.y + v.z * aj4.z + v.w * aj4.w;
    for (int off = 16; off; off >>= 1) {
        si += __shfl_down(si, off, 32);
        sj += __shfl_down(sj, off, 32);
    }
    if (lane == 0) { a_i[g] = si; a_j[g] = sj; }
}

// ---------------------------------------------------------------------------
// K4: per-edge leaky-relu logit + segment max via ordered-uint atomicMax.
// Edges e >= E are the N appended self-loops (src = dst = e - E).
// ---------------------------------------------------------------------------
__global__ void k_alpha_max(const int* __restrict__ src, const int* __restrict__ dstv,
                            const float* __restrict__ a_i, const float* __restrict__ a_j,
                            unsigned* __restrict__ amaxkey, int E, int Et) {
    int e = blockIdx.x * blockDim.x + threadIdx.x;
    if (e >= Et) return;
    int s, d0;
    if (e < E) { s = src[e]; d0 = dstv[e]; } else { s = d0 = e - E; }
    float al = a_i[d0] + a_j[s];
    al = al > 0.0f ? al : NEG_SLOPE * al;
    atomicMax(&amaxkey[d0], f2ord(al));
}

// ---------------------------------------------------------------------------
// K5: per-edge exp(alpha - amax[dst]); stash weight; segment-sum denom.
// ---------------------------------------------------------------------------
__global__ void k_exp_sum(const int* __restrict__ src, const int* __restrict__ dstv,
                          const float* __restrict__ a_i, const float* __restrict__ a_j,
                          const unsigned* __restrict__ amaxkey,
                          float* __restrict__ denom, float* __restrict__ wgt,
                          int E, int Et) {
    int e = blockIdx.x * blockDim.x + threadIdx.x;
    if (e >= Et) return;
    int s, d0;
    if (e < E) { s = src[e]; d0 = dstv[e]; } else { s = d0 = e - E; }
    float al = a_i[d0] + a_j[s];
    al = al > 0.0f ? al : NEG_SLOPE * al;
    float w = expf(al - ord2f(amaxkey[d0]));
    wgt[e] = w;
    atomicAdd(&denom[d0], w);
}

// ---------------------------------------------------------------------------
// K6: message aggregation out[dst] += (w/denom[dst]) * ox[src].
// One wave32 per edge; 4 coalesced float atomics per lane (each instruction's
// 32 lanes cover 128 consecutive bytes). ox/out are L2-resident so gathers
// and atomics stay on-chip.
// ---------------------------------------------------------------------------
__global__ void k_message(const int* __restrict__ src, const int* __restrict__ dstv,
                          const float* __restrict__ ox, const float* __restrict__ wgt,
                          const float* __restrict__ denom, float* __restrict__ out,
                          int E, int Et) {
    int g    = (int)(((long long)blockIdx.x * blockDim.x + threadIdx.x) >> 5);
    int lane = threadIdx.x & 31;
    if (g >= Et) return;
    int s, d0;
    if (g < E) { s = src[g]; d0 = dstv[g]; } else { s = d0 = g - E; }
    float coeff = wgt[g] / (denom[d0] + 1e-16f);
    const float* xr = ox + (long long)s * D;
    float* orow = out + (long long)d0 * D;
#pragma unroll
    for (int i = 0; i < 4; ++i) {
        int c = i * 32 + lane;
        atomicAdd(&orow[c], coeff * xr[c]);
    }
}

// ---------------------------------------------------------------------------
// Launch: inputs are {x, edge_index, node_label, weight_id, att} (fp32/int32).
// Workspace (~32 MB): ox | a_i | a_j | amaxkey | denom | wgt | cnt | lists | Wt
// ---------------------------------------------------------------------------
extern "C" void kernel_launch(void* const* d_in, const int* in_sizes, int n_in,
                              void* d_out, int out_size, void* d_ws, size_t ws_size,
                              hipStream_t stream) {
    const float* x   = (const float*)d_in[0];
    const int*   ei  = (const int*)d_in[1];
    const int*   lab = (const int*)d_in[2];
    const float* W   = (const float*)d_in[3];
    const float* att = (const float*)d_in[4];
    float*       out = (float*)d_out;

    const int N  = in_sizes[0] / D;
    const int E  = in_sizes[1] / 2;
    const int Et = E + N;                 // with appended self-loops
    const int* srcp = ei;
    const int* dstp = ei + E;

    float*    ox      = (float*)d_ws;
    float*    a_i     = ox + (size_t)N * D;
    float*    a_j     = a_i + N;
    unsigned* amaxkey = (unsigned*)(a_j + N);
    float*    denom   = (float*)(amaxkey + N);
    float*    wgt     = denom + N;
    int*      cnt     = (int*)(wgt + Et);
    int*      lists   = cnt + 8;          // 7 segments of N ints
    float*    Wtr     = (float*)(lists + (size_t)NUM_LABELS * N);

    const long long total = (long long)N * D;

    k_init<<<(unsigned)((total + 255) / 256), 256, 0, stream>>>(
        x, ox, out, amaxkey, denom, cnt, N);

    k_wtrans<<<(NUM_LABELS * D * D + 255) / 256, 256, 0, stream>>>(W, Wtr);

    k_bucket<<<(N + 255) / 256, 256, 0, stream>>>(lab, cnt, lists, N);

    dim3 gGemm((N + 31) / 32, NUM_LABELS);
    k_gemm<<<gGemm, 128, 0, stream>>>(x, Wtr, cnt, lists, ox, N);

    k_dots<<<(unsigned)(((long long)N * 32 + 255) / 256), 256, 0, stream>>>(
        ox, att, a_i, a_j, N);

    k_alpha_max<<<(Et + 255) / 256, 256, 0, stream>>>(
        srcp, dstp, a_i, a_j, amaxkey, E, Et);

    k_exp_sum<<<(Et + 255) / 256, 256, 0, stream>>>(
        srcp, dstp, a_i, a_j, amaxkey, denom, wgt, E, Et);

    k_message<<<(unsigned)(((long long)Et * 32 + 255) / 256), 256, 0, stream>>>(
        srcp, dstp, ox, wgt, denom, out, E, Et);
}